// Base_77025943486850
// MI455X (gfx1250) — compile-verified
//
#include <hip/hip_runtime.h>
#include <stdint.h>

#define EPS_F 1e-10f

typedef float v4f __attribute__((ext_vector_type(4)));
typedef int   v4i __attribute__((ext_vector_type(4)));

// ---------------------------------------------------------------------------
// Workspace layout (all float, offsets computed device-side from *pNI):
//   ws[0      .. NI-1 ]  base   (segment sum of ratings)
//   ws[NI     .. 2NI-1]  count  (segment sum of rating>0)
//   ws[2NI+0]            sum of per-item averages over seen items
//   ws[2NI+1]            n_seen
//   ws[2NI+2]            loss accumulator (sum of squared errors)
// ---------------------------------------------------------------------------

__global__ __launch_bounds__(256) void k_zero(float* __restrict__ ws,
                                              const int* __restrict__ pNI) {
  const int NI = *pNI;
  const int n  = 2 * NI + 3;
  const int n4 = n >> 2;
  v4f z = {0.0f, 0.0f, 0.0f, 0.0f};
  v4f* __restrict__ w4 = (v4f*)ws;
  const int gid    = blockIdx.x * blockDim.x + threadIdx.x;
  const int stride = gridDim.x * blockDim.x;
  for (int i = gid; i < n4; i += stride) w4[i] = z;    // global_store_b128
  const int rem = n & 3;
  if (gid < rem) ws[(n4 << 2) + gid] = 0.0f;
}

// Scatter: stage 4 elements/lane via async global->LDS b128 DMA (ASYNCcnt
// path), consume with ds_load_b128, then non-returning f32 atomics into the
// L2-resident base/count arrays.
__global__ __launch_bounds__(256) void k_scatter(const float* __restrict__ rating,
                                                 const int* __restrict__ item,
                                                 float* __restrict__ ws,
                                                 const int* __restrict__ pNI,
                                                 int n) {
  __shared__ __align__(16) float s_rate[256 * 4];
  __shared__ __align__(16) int   s_item[256 * 4];
  const int NI = *pNI;
  float* __restrict__ base = ws;
  float* __restrict__ cnt  = ws + NI;

  const int tid = threadIdx.x;
  // Low 32 bits of a generic LDS pointer = byte offset into the workgroup's
  // LDS allocation (flat-aperture rule) -> VDST of the async-to-LDS op.
  const uint32_t lds_r = (uint32_t)(size_t)&s_rate[tid * 4];
  const uint32_t lds_i = (uint32_t)(size_t)&s_item[tid * 4];

  const int n4     = n >> 2;
  const int gid    = blockIdx.x * blockDim.x + tid;
  const int stride = gridDim.x * blockDim.x;

  for (int i = gid; i < n4; i += stride) {
    if (i + stride < n4) {
      __builtin_prefetch(rating + (size_t)(i + stride) * 4, 0, 1);  // global_prefetch_b8
      __builtin_prefetch(item + (size_t)(i + stride) * 4, 0, 1);
    }
    const float* gr = rating + (size_t)i * 4;
    const int*   gi = item + (size_t)i * 4;
    // Async DMA: 16B per lane into this lane's private LDS slot.
    asm volatile("global_load_async_to_lds_b128 %0, %1, off"
                 :: "v"(lds_r), "v"(gr) : "memory");
    asm volatile("global_load_async_to_lds_b128 %0, %1, off"
                 :: "v"(lds_i), "v"(gi) : "memory");
    asm volatile("s_wait_asynccnt 0x0" ::: "memory");

    const v4f rv = *(const v4f*)&s_rate[tid * 4];   // ds_load_b128
    const v4i iv = *(const v4i*)&s_item[tid * 4];
#pragma unroll
    for (int k = 0; k < 4; ++k) {
      const float r  = rv[k];
      const int   it = iv[k];
      atomicAdd(&base[it], r);                 // non-returning global_atomic_add_f32
      if (r > 0.0f) atomicAdd(&cnt[it], 1.0f);
    }
  }

  // Scalar tail (n not divisible by 4).
  const int rem = n & 3;
  if (rem && gid < rem) {
    const int i = (n4 << 2) + gid;
    const float r  = rating[i];
    const int   it = item[i];
    atomicAdd(&base[it], r);
    if (r > 0.0f) atomicAdd(&cnt[it], 1.0f);
  }
}

// Per-item reduction: sum of (base/count) over seen items and #seen.
__global__ __launch_bounds__(256) void k_itemred(float* __restrict__ ws,
                                                 const int* __restrict__ pNI) {
  __shared__ float s_sum[256];
  __shared__ float s_cnt[256];
  const int NI = *pNI;
  const v4f* __restrict__ base4 = (const v4f*)ws;
  const v4f* __restrict__ cnt4  = (const v4f*)(ws + NI);
  float* __restrict__ acc       = ws + 2 * NI;

  const int tid    = threadIdx.x;
  const int gid    = blockIdx.x * blockDim.x + tid;
  const int stride = gridDim.x * blockDim.x;
  const int n4     = NI >> 2;

  float sum = 0.0f, ns = 0.0f;
  for (int i = gid; i < n4; i += stride) {
    const v4f c = cnt4[i];    // b128 loads; RT hints keep arrays hot in L2
    const v4f b = base4[i];
#pragma unroll
    for (int k = 0; k < 4; ++k)
      if (c[k] != 0.0f) { sum += b[k] / c[k]; ns += 1.0f; }
  }
  const int rem = NI & 3;
  if (rem && gid < rem) {
    const int i = (n4 << 2) + gid;
    const float c = ws[NI + i];
    if (c != 0.0f) { sum += ws[i] / c; ns += 1.0f; }
  }

  s_sum[tid] = sum;
  s_cnt[tid] = ns;
  __syncthreads();
  for (int off = 128; off > 0; off >>= 1) {
    if (tid < off) {
      s_sum[tid] += s_sum[tid + off];
      s_cnt[tid] += s_cnt[tid + off];
    }
    __syncthreads();
  }
  if (tid == 0) {
    atomicAdd(&acc[0], s_sum[0]);
    atomicAdd(&acc[1], s_cnt[0]);
  }
}

// Prediction + MSE partial reduction. Streams use NT b128 accesses; the
// base/count gathers are scalar (random) but L2-resident.
__global__ __launch_bounds__(256) void k_predict(const int* __restrict__ titem,
                                                 const float* __restrict__ trating,
                                                 float* __restrict__ ws,
                                                 const int* __restrict__ pNI,
                                                 float* __restrict__ pred,
                                                 int n) {
  __shared__ float s_loss[256];
  const int NI = *pNI;
  const float* __restrict__ base = ws;
  const float* __restrict__ cnt  = ws + NI;
  float* __restrict__ acc        = ws + 2 * NI;

  const float n_seen = acc[1];
  const float gmean  = acc[0] / fmaxf(n_seen, 1.0f);

  const int tid    = threadIdx.x;
  const int gid    = blockIdx.x * blockDim.x + tid;
  const int stride = gridDim.x * blockDim.x;
  const int n4     = n >> 2;

  const v4i* __restrict__ ti4 = (const v4i*)titem;
  const v4f* __restrict__ tr4 = (const v4f*)trating;
  v4f* __restrict__ pr4       = (v4f*)pred;

  float lsum = 0.0f;
  for (int i = gid; i < n4; i += stride) {
    if (i + stride < n4) {
      __builtin_prefetch(titem + (size_t)(i + stride) * 4, 0, 1);
      __builtin_prefetch(trating + (size_t)(i + stride) * 4, 0, 1);
    }
    const v4i ti = __builtin_nontemporal_load(&ti4[i]);   // global_load_b128 th:NT
    const v4f tr = __builtin_nontemporal_load(&tr4[i]);
    v4f p;
#pragma unroll
    for (int k = 0; k < 4; ++k) {
      const float tb = base[ti[k]];   // L2-resident gather
      const float tc = cnt[ti[k]];
      p[k] = (tc == 0.0f) ? gmean : tb / (tc + EPS_F);
      const float d = p[k] - tr[k];
      lsum += d * d;
    }
    __builtin_nontemporal_store(p, &pr4[i]);              // global_store_b128 th:NT
  }
  const int rem = n & 3;
  if (rem && gid < rem) {
    const int i = (n4 << 2) + gid;
    const int   ti = titem[i];
    const float tr = trating[i];
    const float tc = cnt[ti];
    const float p  = (tc == 0.0f) ? gmean : base[ti] / (tc + EPS_F);
    pred[i] = p;
    const float d = p - tr;
    lsum += d * d;
  }

  s_loss[tid] = lsum;
  __syncthreads();
  for (int off = 128; off > 0; off >>= 1) {
    if (tid < off) s_loss[tid] += s_loss[tid + off];
    __syncthreads();
  }
  if (tid == 0) atomicAdd(&acc[2], s_loss[0]);
}

__global__ void k_final(const float* __restrict__ ws,
                        const int* __restrict__ pNI,
                        float* __restrict__ out_loss, float invT) {
  const int NI = *pNI;
  *out_loss = ws[2 * NI + 2] * invT;
}

extern "C" void kernel_launch(void* const* d_in, const int* in_sizes, int n_in,
                              void* d_out, int out_size, void* d_ws, size_t ws_size,
                              hipStream_t stream) {
  const float* rating  = (const float*)d_in[0];
  const int*   item    = (const int*)d_in[1];
  const int*   titem   = (const int*)d_in[2];
  const float* trating = (const float*)d_in[3];
  const int*   pNI     = (const int*)d_in[4];  // device scalar (num_items)

  const int B = in_sizes[0];
  const int T = in_sizes[2];

  float* ws   = (float*)d_ws;
  float* pred = (float*)d_out;

  k_zero<<<2048, 256, 0, stream>>>(ws, pNI);

  int sb = ((B >> 2) + 255) / 256;
  if (sb < 1) sb = 1;
  if (sb > 16384) sb = 16384;
  k_scatter<<<sb, 256, 0, stream>>>(rating, item, ws, pNI, B);

  k_itemred<<<2048, 256, 0, stream>>>(ws, pNI);

  int pb = ((T >> 2) + 255) / 256;
  if (pb < 1) pb = 1;
  if (pb > 16384) pb = 16384;
  k_predict<<<pb, 256, 0, stream>>>(titem, trating, ws, pNI, pred, T);

  k_final<<<1, 1, 0, stream>>>(ws, pNI, pred + T, 1.0f / (float)T);
}